// SymmetryBreakingPolynomial_90993177133394
// MI455X (gfx1250) — compile-verified
//
#include <hip/hip_runtime.h>
#include <hip/hip_bf16.h>

// ---------------------------------------------------------------------------
// e3nn-style equivariant message passing for MI455X (gfx1250, wave32).
// Per-edge MLP GEMMs run through V_WMMA_F32_16X16X4_F32 (exact fp32 matmul).
// Wigner-3j real tensors computed on-device (double) to mirror the reference.
// ---------------------------------------------------------------------------

typedef __attribute__((ext_vector_type(2))) float v2f;
typedef __attribute__((ext_vector_type(8))) float v8f;

__device__ __forceinline__ v8f wmma_f32_4(v2f a, v2f b, v8f c) {
  // D = A(16x4,f32) x B(4x16,f32) + C(16x16,f32)
  return __builtin_amdgcn_wmma_f32_16x16x4_f32(false, a, false, b, (short)0, c,
                                               false, false);
}

__device__ __forceinline__ float sigmoidf_(float x) { return 1.f / (1.f + expf(-x)); }
__device__ __forceinline__ float susf_(float x) { return x > 0.f ? expf(-1.f / x) : 0.f; }

// ---- Wigner 3j (real basis, e3nn convention), computed in double ----------
__device__ double dfact_(int n) { double r = 1.0; for (int i = 2; i <= n; ++i) r *= (double)i; return r; }

__device__ double dcg_(int l1, int m1, int l2, int m2, int l3, int m3) {
  if (m1 + m2 != m3) return 0.0;
  double pref = sqrt((double)(2 * l3 + 1) * dfact_(l3 + l1 - l2) * dfact_(l3 - l1 + l2) *
                     dfact_(l1 + l2 - l3) / dfact_(l1 + l2 + l3 + 1));
  pref *= sqrt(dfact_(l3 + m3) * dfact_(l3 - m3) * dfact_(l1 - m1) * dfact_(l1 + m1) *
               dfact_(l2 - m2) * dfact_(l2 + m2));
  double s = 0.0;
  for (int k = 0; k <= l1 + l2 - l3; ++k) {
    int d0 = k, d1 = l1 + l2 - l3 - k, d2 = l1 - m1 - k, d3 = l2 + m2 - k,
        d4 = l3 - l2 + m1 + k, d5 = l3 - l1 - m2 + k;
    if (d0 < 0 || d1 < 0 || d2 < 0 || d3 < 0 || d4 < 0 || d5 < 0) continue;
    double den = dfact_(d0) * dfact_(d1) * dfact_(d2) * dfact_(d3) * dfact_(d4) * dfact_(d5);
    s += ((k & 1) ? -1.0 : 1.0) / den;
  }
  return pref * s;
}

__device__ void build_U_(int l, double* Ur, double* Ui) {
  int d = 2 * l + 1;
  for (int i = 0; i < d * d; ++i) { Ur[i] = 0.0; Ui[i] = 0.0; }
  Ur[l * d + l] = 1.0;
  const double is2 = 1.0 / sqrt(2.0);
  for (int m = 1; m <= l; ++m) {
    double sgn = (m & 1) ? -1.0 : 1.0;
    Ur[(l + m) * d + (l + m)] = sgn * is2;
    Ur[(l + m) * d + (l - m)] = is2;
    Ui[(l - m) * d + (l - m)] = is2;
    Ui[(l - m) * d + (l + m)] = -sgn * is2;
  }
}

__device__ void w3j_real_(int l1, int l2, int l3, float* out, double scale) {
  int d1 = 2 * l1 + 1, d2 = 2 * l2 + 1, d3 = 2 * l3 + 1;
  double Wc[7 * 7 * 7];
  for (int i = 0; i < d1 * d2 * d3; ++i) Wc[i] = 0.0;
  for (int m1 = -l1; m1 <= l1; ++m1)
    for (int m2 = -l2; m2 <= l2; ++m2) {
      int m3 = -(m1 + m2);
      if (m3 < -l3 || m3 > l3) continue;
      int ex = l1 - l2 - m3;
      double sgn = ((((ex % 2) + 2) % 2) == 0) ? 1.0 : -1.0;
      Wc[((m1 + l1) * d2 + (m2 + l2)) * d3 + (m3 + l3)] =
          sgn / sqrt((double)(2 * l3 + 1)) * dcg_(l1, m1, l2, m2, l3, -m3);
    }
  double U1r[49], U1i[49], U2r[49], U2i[49], U3r[49], U3i[49];
  build_U_(l1, U1r, U1i); build_U_(l2, U2r, U2i); build_U_(l3, U3r, U3i);
  double Tre[7 * 7 * 7], Tim[7 * 7 * 7];
  double mre = 0.0, mim = 0.0;
  for (int a = 0; a < d1; ++a)
    for (int b = 0; b < d2; ++b)
      for (int c = 0; c < d3; ++c) {
        double re = 0.0, im = 0.0;
        for (int m = 0; m < d1; ++m)
          for (int n = 0; n < d2; ++n)
            for (int k = 0; k < d3; ++k) {
              double w = Wc[(m * d2 + n) * d3 + k];
              if (w == 0.0) continue;
              double r1 = U1r[a * d1 + m], i1 = U1i[a * d1 + m];
              double r2 = U2r[b * d2 + n], i2 = U2i[b * d2 + n];
              double r3 = U3r[c * d3 + k], i3 = U3i[c * d3 + k];
              double rr = r1 * r2 - i1 * i2, ri = r1 * i2 + i1 * r2;
              double pr = rr * r3 - ri * i3, pi = rr * i3 + ri * r3;
              re += pr * w; im += pi * w;
            }
        Tre[(a * d2 + b) * d3 + c] = re; Tim[(a * d2 + b) * d3 + c] = im;
        if (fabs(re) > mre) mre = fabs(re);
        if (fabs(im) > mim) mim = fabs(im);
      }
  double* pick = (mre >= mim) ? Tre : Tim;
  double nrm = 0.0;
  for (int i = 0; i < d1 * d2 * d3; ++i) nrm += pick[i] * pick[i];
  nrm = sqrt(nrm);
  double inv = (nrm > 0.0) ? (scale / nrm) : 0.0;
  for (int i = 0; i < d1 * d2 * d3; ++i) out[i] = (float)(pick[i] * inv);
}

// const-workspace offsets (floats); tensors already scaled by sqrt(2*lo+1)
#define CW000 0
#define CW011 1
#define CW022 10
#define CW101 35
#define CW110 44
#define CW112 53
#define CW121 98
#define CW132 143
#define CW202 248
#define CW222 273
#define CWTOT 398

__global__ void zero_kernel(float* node, long long nodeElems, float* out5) {
  long long idx = (long long)blockIdx.x * blockDim.x + threadIdx.x;
  long long stride = (long long)gridDim.x * blockDim.x;
  for (long long j = idx; j < nodeElems; j += stride) node[j] = 0.f;
  if (idx < 5) out5[idx] = 0.f;
}

__global__ void w3j_init_kernel(float* cw) {
  if (blockIdx.x != 0 || threadIdx.x != 0) return;
  const double s3 = 1.7320508075688772, s5 = 2.2360679774997896;
  w3j_real_(0, 0, 0, cw + CW000, 1.0);
  w3j_real_(0, 1, 1, cw + CW011, s3);
  w3j_real_(0, 2, 2, cw + CW022, s5);
  w3j_real_(1, 0, 1, cw + CW101, s3);
  w3j_real_(1, 1, 0, cw + CW110, 1.0);
  w3j_real_(1, 1, 2, cw + CW112, s5);
  w3j_real_(1, 2, 1, cw + CW121, s3);
  w3j_real_(1, 3, 2, cw + CW132, s5);
  w3j_real_(2, 0, 2, cw + CW202, s5);
  w3j_real_(2, 2, 2, cw + CW222, s5);
}

// ---------------------------------------------------------------------------
// Stage 1: per-edge SH + radial embedding + MLP (WMMA f32) + TP1 + scatter
// block = 64 threads (2 waves); each wave runs 2 WMMA M-tiles (32 edges).
// ---------------------------------------------------------------------------
__global__ __launch_bounds__(64) void edge_stage1(
    const float* __restrict__ pos, const float* __restrict__ features,
    const float* __restrict__ sb, const float* __restrict__ w1,
    const float* __restrict__ w2, const float* __restrict__ cw,
    const int* __restrict__ efrom, const int* __restrict__ eto,
    float* __restrict__ node, int E) {
  __shared__ float embT[64][20];  // scaled embedding (A of GEMM1)
  __shared__ float hT[64][52];    // hidden acts / sqrt(50) (A of GEMM2), K-padded
  __shared__ float wT[64][40];    // per-edge TP weights

  const int tid = threadIdx.x;
  const int e = blockIdx.x * 64 + tid;
  const bool valid = (e < E);

  int nf = 0, nt = 0;
  float sh[16];
#pragma unroll
  for (int i = 0; i < 16; ++i) sh[i] = 0.f;

  if (valid) {
    nf = efrom[e]; nt = eto[e];
    float vx = pos[3 * nt + 0] - pos[3 * nf + 0];
    float vy = pos[3 * nt + 1] - pos[3 * nf + 1];
    float vz = pos[3 * nt + 2] - pos[3 * nf + 2];
    float dist = sqrtf(vx * vx + vy * vy + vz * vz + 1e-12f);
    float inv = 1.f / dist;
    float x = vx * inv, y = vy * inv, z = vz * inv;
    const float s3 = 1.7320508075688772f, s5 = 2.23606797749979f, s15 = 3.872983346207417f;
    sh[0] = 1.f;
    sh[1] = s3 * y; sh[2] = s3 * z; sh[3] = s3 * x;
    sh[4] = s15 * x * y; sh[5] = s15 * y * z;
    sh[6] = 0.5f * s5 * (3.f * z * z - 1.f);
    sh[7] = s15 * x * z; sh[8] = 0.5f * s15 * (x * x - y * y);
    sh[9]  = 2.0916500663351889f * y * (3.f * x * x - y * y);
    sh[10] = 10.246950765959598f * x * y * z;
    sh[11] = 1.6201851746019651f * y * (5.f * z * z - 1.f);
    sh[12] = 1.3228756555322954f * z * (5.f * z * z - 3.f);
    sh[13] = 1.6201851746019651f * x * (5.f * z * z - 1.f);
    sh[14] = 5.1234753829797990f * z * (x * x - y * y);
    sh[15] = 2.0916500663351889f * x * (3.f * x * x - y * y);
    const float step = 2.f / 21.f;
#pragma unroll
    for (int i = 0; i < 20; ++i) {
      float diff = (dist - (float)(i + 1) * step) / step;
      float a = susf_(diff + 1.f), b = susf_(1.f - diff);
      // 1.14136*e^2 * sus*sus, folded with 1/sqrt(20) (w1 normalization)
      embT[tid][i] = 8.433573069f * a * b * 0.22360679775f;
    }
  } else {
#pragma unroll
    for (int i = 0; i < 20; ++i) embT[tid][i] = 0.f;
  }
  __syncthreads();

  const int lane = tid & 31;
  const int wv = tid >> 5;
  const int lh = lane >> 4;   // half-wave select (K offset +2)
  const int l15 = lane & 15;
  const int r0 = wv * 32;

  // ---- GEMM1: (32x20) @ (20x50) via v_wmma_f32_16x16x4_f32 ----
  for (int t = 0; t < 4; ++t) {
    v8f c0 = {}; v8f c1 = {};
    const int n = t * 16 + l15;
#pragma unroll
    for (int s = 0; s < 5; ++s) {
      const int k0 = 4 * s + 2 * lh;
      v2f a0, a1, b;
      a0[0] = embT[r0 + l15][k0];      a0[1] = embT[r0 + l15][k0 + 1];
      a1[0] = embT[r0 + 16 + l15][k0]; a1[1] = embT[r0 + 16 + l15][k0 + 1];
      b[0] = (n < 50) ? w1[k0 * 50 + n] : 0.f;
      b[1] = (n < 50) ? w1[(k0 + 1) * 50 + n] : 0.f;
      c0 = wmma_f32_4(a0, b, c0);
      c1 = wmma_f32_4(a1, b, c1);
    }
#pragma unroll
    for (int v = 0; v < 8; ++v) {
      if (n < 52) {  // SILU_NORM * silu(y), folded with 1/sqrt(50)
        float y0 = c0[v], y1 = c1[v];
        hT[r0 + v + 8 * lh][n]      = 1.679177f * y0 / (1.f + expf(-y0)) * 0.14142135623730951f;
        hT[r0 + 16 + v + 8 * lh][n] = 1.679177f * y1 / (1.f + expf(-y1)) * 0.14142135623730951f;
      }
    }
  }
  __syncthreads();

  // ---- GEMM2: (32x50) @ (50x40), K padded to 52 ----
  for (int t = 0; t < 3; ++t) {
    v8f c0 = {}; v8f c1 = {};
    const int n = t * 16 + l15;
#pragma unroll
    for (int s = 0; s < 13; ++s) {
      const int k0 = 4 * s + 2 * lh;
      v2f a0, a1, b;
      a0[0] = hT[r0 + l15][k0];      a0[1] = hT[r0 + l15][k0 + 1];
      a1[0] = hT[r0 + 16 + l15][k0]; a1[1] = hT[r0 + 16 + l15][k0 + 1];
      b[0] = (k0 < 50 && n < 40) ? w2[k0 * 40 + n] : 0.f;
      b[1] = (k0 + 1 < 50 && n < 40) ? w2[(k0 + 1) * 40 + n] : 0.f;
      c0 = wmma_f32_4(a0, b, c0);
      c1 = wmma_f32_4(a1, b, c1);
    }
#pragma unroll
    for (int v = 0; v < 8; ++v) {
      if (n < 40) {
        wT[r0 + v + 8 * lh][n] = c0[v];
        wT[r0 + 16 + v + 8 * lh][n] = c1[v];
      }
    }
  }
  __syncthreads();

  // ---- TP1: 8 CG paths, scatter-add into node[edge_from] ----
  if (valid) {
    float w[40];
#pragma unroll
    for (int i = 0; i < 40; ++i) w[i] = wT[tid][i];
    const float s0 = features[nt];
    const float vv0 = sb[0], vv1 = sb[1], vv2 = sb[2];
    const float vv[3] = {vv0, vv1, vv2};
    const float* Y1 = sh + 1;
    const float* Y2 = sh + 4;
    const float* Y3 = sh + 9;
    const float w000 = cw[CW000];
    float p2[3], p3[5], p4[3], t6[5], t7[3], t8[5];
#pragma unroll
    for (int k = 0; k < 3; ++k) { float a = 0; for (int j = 0; j < 3; ++j) a += cw[CW011 + j * 3 + k] * Y1[j]; p2[k] = a; }
#pragma unroll
    for (int k = 0; k < 5; ++k) { float a = 0; for (int j = 0; j < 5; ++j) a += cw[CW022 + j * 5 + k] * Y2[j]; p3[k] = a; }
#pragma unroll
    for (int k = 0; k < 3; ++k) { float a = 0; for (int i = 0; i < 3; ++i) a += cw[CW101 + i * 3 + k] * vv[i]; p4[k] = a; }
    float p5 = 0.f;
    for (int i = 0; i < 3; ++i) for (int j = 0; j < 3; ++j) p5 += cw[CW110 + i * 3 + j] * vv[i] * Y1[j];
#pragma unroll
    for (int k = 0; k < 5; ++k) { float a = 0; for (int i = 0; i < 3; ++i) for (int j = 0; j < 3; ++j) a += cw[CW112 + (i * 3 + j) * 5 + k] * vv[i] * Y1[j]; t6[k] = a; }
#pragma unroll
    for (int k = 0; k < 3; ++k) { float a = 0; for (int i = 0; i < 3; ++i) for (int j = 0; j < 5; ++j) a += cw[CW121 + (i * 5 + j) * 3 + k] * vv[i] * Y2[j]; t7[k] = a; }
#pragma unroll
    for (int k = 0; k < 5; ++k) { float a = 0; for (int i = 0; i < 3; ++i) for (int j = 0; j < 7; ++j) a += cw[CW132 + (i * 7 + j) * 5 + k] * vv[i] * Y3[j]; t8[k] = a; }
    const float inv_s2 = 0.70710678118654752f;  // 1/sqrt(fan=2)
    const float inv_s3 = 0.57735026918962576f;  // 1/sqrt(fan=3)
    float* nb = node + (size_t)nf * 50;
#pragma unroll
    for (int u = 0; u < 5; ++u) {
      atomicAdd(&nb[u], (w000 * s0 * w[u] + p5 * w[20 + u]) * inv_s2);
#pragma unroll
      for (int k = 0; k < 3; ++k)
        atomicAdd(&nb[10 + 3 * u + k],
                  (s0 * p2[k] * w[5 + u] + p4[k] * w[15 + u] + t7[k] * w[30 + u]) * inv_s3);
#pragma unroll
      for (int k = 0; k < 5; ++k)
        atomicAdd(&nb[25 + 5 * u + k],
                  (s0 * p3[k] * w[10 + u] + t6[k] * w[25 + u] + t8[k] * w[35 + u]) * inv_s3);
    }
  }
}

// ---------------------------------------------------------------------------
// Stage 2: per-node norm-gated activation over MID irreps
// ---------------------------------------------------------------------------
__global__ void norm_act_kernel(const float* __restrict__ node,
                                const float* __restrict__ bias,
                                float* __restrict__ nodeact, int N) {
  int i = blockIdx.x * blockDim.x + threadIdx.x;
  if (i >= N) return;
  const float* f = node + (size_t)i * 50;
  float* o = nodeact + (size_t)i * 50;
#pragma unroll
  for (int u = 0; u < 5; ++u) {  // 5x0e
    float x = f[u];
    float nrm = sqrtf(x * x + 1e-12f);
    o[u] = x * (sigmoidf_(nrm + bias[u]) / nrm);
  }
#pragma unroll
  for (int u = 0; u < 5; ++u) {  // 5x0o (zero channels)
    float x = f[5 + u];
    float nrm = sqrtf(x * x + 1e-12f);
    o[5 + u] = x * (sigmoidf_(nrm + bias[5 + u]) / nrm);
  }
#pragma unroll
  for (int u = 0; u < 5; ++u) {  // 5x1o
    float a = f[10 + 3 * u], b = f[10 + 3 * u + 1], c = f[10 + 3 * u + 2];
    float nrm = sqrtf(a * a + b * b + c * c + 1e-12f);
    float s = sigmoidf_(nrm + bias[10 + u]) / nrm;
    o[10 + 3 * u] = a * s; o[10 + 3 * u + 1] = b * s; o[10 + 3 * u + 2] = c * s;
  }
#pragma unroll
  for (int u = 0; u < 5; ++u) {  // 5x2e
    float acc = 1e-12f;
    for (int k = 0; k < 5; ++k) { float x = f[25 + 5 * u + k]; acc += x * x; }
    float nrm = sqrtf(acc);
    float s = sigmoidf_(nrm + bias[15 + u]) / nrm;
    for (int k = 0; k < 5; ++k) o[25 + 5 * u + k] = f[25 + 5 * u + k] * s;
  }
}

// ---------------------------------------------------------------------------
// Stage 3: per-edge TP2 (-> single 2e irrep) + global reduction to 5 floats
// ---------------------------------------------------------------------------
__global__ __launch_bounds__(64) void edge_stage2(
    const float* __restrict__ pos, const float* __restrict__ nodeact,
    const float* __restrict__ tp2w, const float* __restrict__ cw,
    const int* __restrict__ efrom, const int* __restrict__ eto,
    float* __restrict__ out, int E) {
  __shared__ float red[5];
  if (threadIdx.x < 5) red[threadIdx.x] = 0.f;
  __syncthreads();
  int e = blockIdx.x * 64 + threadIdx.x;
  if (e < E) {
    int nf = efrom[e], nt = eto[e];
    float vx = pos[3 * nt + 0] - pos[3 * nf + 0];
    float vy = pos[3 * nt + 1] - pos[3 * nf + 1];
    float vz = pos[3 * nt + 2] - pos[3 * nf + 2];
    float dist = sqrtf(vx * vx + vy * vy + vz * vz + 1e-12f);
    float inv = 1.f / dist;
    float x = vx * inv, y = vy * inv, z = vz * inv;
    const float s3 = 1.7320508075688772f, s5 = 2.23606797749979f, s15 = 3.872983346207417f;
    float Y1[3] = {s3 * y, s3 * z, s3 * x};
    float Y2[5] = {s15 * x * y, s15 * y * z, 0.5f * s5 * (3.f * z * z - 1.f),
                   s15 * x * z, 0.5f * s15 * (x * x - y * y)};
    float Y3[7] = {2.0916500663351889f * y * (3.f * x * x - y * y),
                   10.246950765959598f * x * y * z,
                   1.6201851746019651f * y * (5.f * z * z - 1.f),
                   1.3228756555322954f * z * (5.f * z * z - 3.f),
                   1.6201851746019651f * x * (5.f * z * z - 1.f),
                   5.1234753829797990f * z * (x * x - y * y),
                   2.0916500663351889f * x * (3.f * x * x - y * y)};
    const float* na = nodeact + (size_t)nt * 50;
    float A0s = 0.f;
    for (int u = 0; u < 5; ++u) A0s += tp2w[u] * na[u];
    float A1[3] = {0, 0, 0}, B1[3] = {0, 0, 0};
    for (int u = 0; u < 5; ++u)
      for (int i = 0; i < 3; ++i) {
        A1[i] += tp2w[5 + u] * na[10 + 3 * u + i];
        B1[i] += tp2w[10 + u] * na[10 + 3 * u + i];
      }
    float A3[5] = {0, 0, 0, 0, 0}, A2[5] = {0, 0, 0, 0, 0};
    for (int u = 0; u < 5; ++u)
      for (int i = 0; i < 5; ++i) {
        A3[i] += tp2w[15 + u] * na[25 + 5 * u + i];
        A2[i] += tp2w[20 + u] * na[25 + 5 * u + i];
      }
#pragma unroll
    for (int k = 0; k < 5; ++k) {
      float acc = 0.f;
      for (int j = 0; j < 5; ++j) acc += cw[CW022 + j * 5 + k] * Y2[j] * A0s;           // (0,2,0)
      for (int i = 0; i < 3; ++i) for (int j = 0; j < 3; ++j) acc += cw[CW112 + (i * 3 + j) * 5 + k] * A1[i] * Y1[j];  // (2,1,0)
      for (int i = 0; i < 3; ++i) for (int j = 0; j < 7; ++j) acc += cw[CW132 + (i * 7 + j) * 5 + k] * B1[i] * Y3[j];  // (2,3,0)
      for (int i = 0; i < 5; ++i) acc += cw[CW202 + i * 5 + k] * A3[i];                  // (3,0,0)
      for (int i = 0; i < 5; ++i) for (int j = 0; j < 5; ++j) acc += cw[CW222 + (i * 5 + j) * 5 + k] * A2[i] * Y2[j];  // (3,2,0)
      atomicAdd(&red[k], acc * 0.2f);  // 1/sqrt(fan=25)
    }
  }
  __syncthreads();
  if (threadIdx.x < 5) atomicAdd(&out[threadIdx.x], red[threadIdx.x]);
}

// ---------------------------------------------------------------------------
extern "C" void kernel_launch(void* const* d_in, const int* in_sizes, int n_in,
                              void* d_out, int out_size, void* d_ws, size_t ws_size,
                              hipStream_t stream) {
  (void)n_in; (void)out_size; (void)ws_size;
  const float* pos      = (const float*)d_in[0];
  const float* features = (const float*)d_in[1];
  const float* sb       = (const float*)d_in[2];
  const float* w1       = (const float*)d_in[3];
  const float* w2       = (const float*)d_in[4];
  const float* tp2w     = (const float*)d_in[5];
  const float* nbias    = (const float*)d_in[6];
  const int*   efrom    = (const int*)d_in[7];
  const int*   eto      = (const int*)d_in[8];
  const int N = in_sizes[1];   // features is (N,1)
  const int E = in_sizes[7];

  float* node    = (float*)d_ws;                 // N*50
  float* nodeact = node + (size_t)N * 50;        // N*50
  float* cw      = nodeact + (size_t)N * 50;     // CWTOT consts
  float* out     = (float*)d_out;                // 5 floats

  zero_kernel<<<1024, 256, 0, stream>>>(node, (long long)N * 50, out);
  w3j_init_kernel<<<1, 32, 0, stream>>>(cw);
  const int gE = (E + 63) / 64;
  edge_stage1<<<gE, 64, 0, stream>>>(pos, features, sb, w1, w2, cw, efrom, eto, node, E);
  norm_act_kernel<<<(N + 255) / 256, 256, 0, stream>>>(node, nbias, nodeact, N);
  edge_stage2<<<gE, 64, 0, stream>>>(pos, nodeact, tp2w, cw, efrom, eto, out, E);
}